// TensorTrainDense_86895778333372
// MI455X (gfx1250) — compile-verified
//
#include <hip/hip_runtime.h>

typedef __attribute__((ext_vector_type(2))) float v2f;
typedef __attribute__((ext_vector_type(4))) float f4;
typedef __attribute__((ext_vector_type(8))) float v8f;
typedef __attribute__((ext_vector_type(4))) int v4i;

// ---------------------------------------------------------------------------
// TT core offsets inside the flat kernel vector (floats):
//   core3 (8 x 256)   @ 0
//   core2 (256 x 256) @ 2048
//   core1 (256 x 256) @ 67584
//   core0 (256 x 8)   @ 133120
//
// Collapsed weight:
//   W[(a c d e),(o0 q1 q2 q)] = sum_{p,p1,p2}
//       core0[a*32+p1, o0] * core1[c*32+p2, p1*8+q1]
//     * core2[d*32+p,  p2*8+q2] * core3[e, p*8+q]
// Stored K-pair interleaved for WMMA B-fragment loads:
//   Wp[(j>>1)*8192 + o*2 + (j&1)]   (j = contraction index, o = output col)
// ---------------------------------------------------------------------------

__global__ __launch_bounds__(256) void tt_build_T2(const float* __restrict__ k,
                                                   float* __restrict__ T2) {
  int tid = blockIdx.x * 256 + threadIdx.x;           // 131072 threads
  int p2 = tid & 31, q1 = (tid >> 5) & 7, c = (tid >> 8) & 7;
  int o0 = (tid >> 11) & 7, a = (tid >> 14) & 7;
  const float* core0 = k + 133120;
  const float* core1 = k + 67584;
  float acc = 0.f;
#pragma unroll
  for (int p1 = 0; p1 < 32; ++p1)
    acc += core0[(a * 32 + p1) * 8 + o0] * core1[(c * 32 + p2) * 256 + p1 * 8 + q1];
  T2[tid] = acc;   // layout [a][o0][c][q1][p2]
}

__global__ __launch_bounds__(256) void tt_build_T3(const float* __restrict__ k,
                                                   const float* __restrict__ T2,
                                                   float* __restrict__ T3) {
  int tid = blockIdx.x * 256 + threadIdx.x;           // 8388608 threads
  int p = tid & 31, q2 = (tid >> 5) & 7, d = (tid >> 8) & 7;
  int q1 = (tid >> 11) & 7, c = (tid >> 14) & 7, o0 = (tid >> 17) & 7, a = (tid >> 20) & 7;
  const float* core2 = k + 2048;
  const float* t2row = T2 + ((((a * 8 + o0) * 8 + c) * 8 + q1) * 32);
  float acc = 0.f;
#pragma unroll
  for (int p2 = 0; p2 < 32; ++p2)
    acc += t2row[p2] * core2[(d * 32 + p) * 256 + p2 * 8 + q2];
  T3[tid] = acc;   // layout [a][o0][c][q1][d][q2][p]
}

__global__ __launch_bounds__(256) void tt_build_W(const float* __restrict__ k,
                                                  const float* __restrict__ T3,
                                                  float* __restrict__ Wp) {
  int tid = blockIdx.x * 256 + threadIdx.x;           // 16777216 threads
  int o = tid & 4095, j = tid >> 12;
  int a = (j >> 9) & 7, c = (j >> 6) & 7, d = (j >> 3) & 7, e = j & 7;
  int o0 = (o >> 9) & 7, q1 = (o >> 6) & 7, q2 = (o >> 3) & 7, q = o & 7;
  const float* core3 = k;                              // 8 x 256
  const float* t3row =
      T3 + ((size_t)((((((a * 8 + o0) * 8 + c) * 8 + q1) * 8 + d) * 8 + q2)) * 32);
  float acc = 0.f;
#pragma unroll
  for (int p = 0; p < 32; ++p)
    acc += t3row[p] * core3[e * 256 + p * 8 + q];
  // K-pair interleaved store: pair row j>>1, column o, slot j&1
  Wp[(size_t)(j >> 1) * 8192 + o * 2 + (j & 1)] = acc;
}

// ---------------------------------------------------------------------------
// Y = X (4096x4096) * W + bias, fp32 via V_WMMA_F32_16X16X4_F32.
// Block: 128x128 C tile, 8 waves as 4(M) x 2(N); each wave owns 32x64.
// K tiles of 32 staged through double-buffered LDS, async when available.
// ---------------------------------------------------------------------------
#define XS_STRIDE 36    // floats per X row in LDS (144B, 16B-aligned)
#define BS_STRIDE 288   // floats per W pair-row in LDS (1152B; 288 % 64 == 32)

#if defined(__gfx1250__) && __has_builtin(__builtin_amdgcn_global_load_async_to_lds_b128)
#define HAVE_ASYNC 1
#else
#define HAVE_ASYNC 0
#endif

#if HAVE_ASYNC
__device__ __forceinline__ void async_copy16(const void* g, void* l) {
  __builtin_amdgcn_global_load_async_to_lds_b128(
      (__attribute__((address_space(1))) v4i*)g,
      (__attribute__((address_space(3))) v4i*)l, 0, 0);
}
__device__ __forceinline__ void wait_async0() {
#if __has_builtin(__builtin_amdgcn_s_wait_asynccnt)
  __builtin_amdgcn_s_wait_asynccnt(0);
#else
  asm volatile("s_wait_asynccnt 0x0" ::: "memory");
#endif
}
#endif

__global__ __launch_bounds__(256) void tt_gemm_wmma(const float* __restrict__ X,
                                                    const float* __restrict__ Wp,
                                                    const float* __restrict__ bias,
                                                    float* __restrict__ Y) {
  __shared__ float sX[2][128 * XS_STRIDE];   // 2 x 18432B
  __shared__ float sB[2][16 * BS_STRIDE];    // 2 x 18432B

  const int tid = threadIdx.x;
  const int lane = tid & 31;
  const int wave = tid >> 5;           // 0..7
  const int mr = lane & 15;
  const int g = lane >> 4;
  const int m0 = blockIdx.y * 128;
  const int n0 = blockIdx.x * 128;
  const int wm = (wave >> 1) * 32;     // wave M origin in tile
  const int wn = (wave & 1) * 64;      // wave N origin in tile

  v8f acc[2][4];
#pragma unroll
  for (int mt = 0; mt < 2; ++mt)
#pragma unroll
    for (int nt = 0; nt < 4; ++nt) acc[mt][nt] = (v8f)0.f;

  // --- tile staging: X 128x32 (1024 16B chunks), W' 16x256 (1024 chunks) ---
  auto stage = [&](int kt, float* xbuf, float* bbuf) {
#pragma unroll
    for (int i = 0; i < 4; ++i) {
      int s = tid + i * 256;
      int r = s >> 3, c = (s & 7) * 4;
      const float* gp = X + (size_t)(m0 + r) * 4096 + kt + c;
      float* lp = xbuf + r * XS_STRIDE + c;
#if HAVE_ASYNC
      async_copy16(gp, lp);
#else
      *(f4*)lp = *(const f4*)gp;
#endif
    }
#pragma unroll
    for (int i = 0; i < 4; ++i) {
      int s = tid + i * 256;
      int kp = s >> 6, c = (s & 63) * 4;
      const float* gp = Wp + (size_t)((kt >> 1) + kp) * 8192 + n0 * 2 + c;
      float* lp = bbuf + kp * BS_STRIDE + c;
#if HAVE_ASYNC
      async_copy16(gp, lp);
#else
      *(f4*)lp = *(const f4*)gp;
#endif
    }
  };

  int cur = 0;
  stage(0, sX[0], sB[0]);
#if HAVE_ASYNC
  wait_async0();
#endif
  __syncthreads();

  for (int kt = 0; kt < 4096; kt += 32) {
    if (kt + 32 < 4096) stage(kt + 32, sX[cur ^ 1], sB[cur ^ 1]);

    const float* xbuf = sX[cur];
    const float* bbuf = sB[cur];
#pragma unroll
    for (int ks = 0; ks < 32; ks += 4) {
      // A fragments 16x4: lane group g holds K = ks+2g, ks+2g+1 (8B aligned)
      v2f a0 = *(const v2f*)&xbuf[(wm + mr) * XS_STRIDE + ks + 2 * g];
      v2f a1 = *(const v2f*)&xbuf[(wm + 16 + mr) * XS_STRIDE + ks + 2 * g];
#pragma unroll
      for (int nt = 0; nt < 4; ++nt) {
        // B fragment: interleaved pair (K=ks+2g, ks+2g+1) at column — one b64
        v2f b = *(const v2f*)&bbuf[((ks >> 1) + g) * BS_STRIDE + (wn + nt * 16 + mr) * 2];
        acc[0][nt] = __builtin_amdgcn_wmma_f32_16x16x4_f32(
            false, a0, false, b, (short)0, acc[0][nt], false, false);
        acc[1][nt] = __builtin_amdgcn_wmma_f32_16x16x4_f32(
            false, a1, false, b, (short)0, acc[1][nt], false, false);
      }
    }
#if HAVE_ASYNC
    wait_async0();
#endif
    __syncthreads();
    cur ^= 1;
  }

  // Epilogue: D layout — VGPR v, lane group g => M = v + 8g ; N = lane&15
#pragma unroll
  for (int mt = 0; mt < 2; ++mt)
#pragma unroll
    for (int nt = 0; nt < 4; ++nt) {
      int col = n0 + wn + nt * 16 + mr;
      float bcol = bias[col];
      int row = m0 + wm + mt * 16 + 8 * g;
#pragma unroll
      for (int v = 0; v < 8; ++v)
        Y[(size_t)(row + v) * 4096 + col] = acc[mt][nt][v] + bcol;
    }
}

extern "C" void kernel_launch(void* const* d_in, const int* in_sizes, int n_in,
                              void* d_out, int out_size, void* d_ws, size_t ws_size,
                              hipStream_t stream) {
  const float* x = (const float*)d_in[0];      // 4096 x 4096
  const float* kern = (const float*)d_in[1];   // 135168
  const float* bias = (const float*)d_in[2];   // 4096
  float* out = (float*)d_out;                  // 4096 x 4096

  // Workspace layout (floats): Wp | T3 | T2  => ~96.5 MB total
  float* Wp = (float*)d_ws;                    // 16777216 (pair-interleaved)
  float* T3 = Wp + 16777216;                   // 8388608
  float* T2 = T3 + 8388608;                    // 131072

  tt_build_T2<<<131072 / 256, 256, 0, stream>>>(kern, T2);
  tt_build_T3<<<8388608 / 256, 256, 0, stream>>>(kern, T2, T3);
  tt_build_W<<<16777216 / 256, 256, 0, stream>>>(kern, T3, Wp);

  dim3 grid(32, 32);   // N tiles, M tiles
  tt_gemm_wmma<<<grid, 256, 0, stream>>>(x, Wp, bias, out);
}